// SpikingLayer_51694226374923
// MI455X (gfx1250) — compile-verified
//
#include <hip/hip_runtime.h>
#include <stdint.h>

// ---------------------------------------------------------------------------
// LIF spiking layer, MI455X (gfx1250).
// Pure HBM streaming: 128 MB read + 64 MB write -> ~8.2 us floor @ 23.3 TB/s.
// 65536 independent (batch,neuron) lanes; sequential recurrence over T=256.
// Input streamed with CDNA5 async global->LDS copies (ASYNCcnt), DEPTH=16
// time-step ring per thread (~8 MB in flight chip-wide). GVS addressing:
// uniform SGPR base advances per step (SALU), per-lane VGPR offsets are
// loop-invariant -> zero per-issue VALU address math. Each thread reads back
// only its own LDS words -> no barriers, only per-wave s_wait_asynccnt.
// ---------------------------------------------------------------------------

#define BATCH   32
#define TSTEPS  256
#define NSYN    2
#define NNEUR   2048
#define TPB     256          // threads per block == neurons per block
#define DEPTH   16           // pipelined time steps (power of two, 2*DEPTH<=63)
#define NGROUP  (TSTEPS / DEPTH)
#define STEPB   ((uint64_t)NSYN * NNEUR * 4)   // 16 KB per time step (input)

// exp(-1/10), exp(-1/5), exp(-1/10)
#define ALPHA_MEM  0.90483741803595957f
#define ALPHA_S0   0.81873075307798182f
#define ALPHA_S1   0.90483741803595957f

// Low 32 bits of a flat pointer into the LDS aperture == group-relative LDS
// byte offset.
__device__ __forceinline__ uint32_t lds_byte_off(const void* p) {
    return (uint32_t)(uintptr_t)p;
}

// GLOBAL_LOAD_ASYNC_TO_LDS_B32, GVS mode:
//   each lane copies 4 bytes from global address sbase plus its voff to the LDS address in lds_off
// Input is streamed once -> non-temporal hint.
__device__ __forceinline__ void async_load_b32(uint32_t lds_off, uint32_t voff,
                                               uint64_t sbase) {
    asm volatile("global_load_async_to_lds_b32 %0, %1, %2 th:TH_LOAD_NT"
                 :: "v"(lds_off), "v"(voff), "s"(sbase)
                 : "memory");
}

// Wait until at most N async ops outstanding.
template <int N>
__device__ __forceinline__ void wait_async_le() {
    asm volatile("s_wait_asynccnt %0" :: "i"(N) : "memory");
}

__global__ __launch_bounds__(TPB) void snn_lif_kernel(const float* __restrict__ in,
                                                      float* __restrict__ out) {
    // Ring: DEPTH slots x (per-thread adjacent pair of synapse words).
    // Readback is one ds_load_b64 per step; async writes hit distinct banks.
    __shared__ float lds[DEPTH][TPB][NSYN];                 // 32 KB

    const int tid = threadIdx.x;
    const int b   = (int)(blockIdx.x >> 3);                 // 8 blocks / batch
    const int n0  = ((int)(blockIdx.x & 7)) * TPB + tid;    // this thread's neuron

    // input:  in[((b*T + t)*NSYN + s)*NNEUR + n]   output: out[(b*T + t)*NNEUR + n]
    // Uniform byte base for (b, t=0, s=0, n=0); per-lane offsets cover (s, n).
    uint64_t base = (uint64_t)(uintptr_t)in + (uint64_t)b * TSTEPS * STEPB;
    const uint32_t off_s0 = (uint32_t)n0 * 4u;              // synapse 0, neuron n0
    const uint32_t off_s1 = off_s0 + (uint32_t)NNEUR * 4u;  // synapse 1, neuron n0

    float* Q = out + ((size_t)b * TSTEPS) * NNEUR + n0;

    const uint32_t l0    = lds_byte_off(&lds[0][tid][0]);
    const uint32_t lslot = TPB * NSYN * 4;                  // 2 KB per depth slot

    // ---- prologue: fill pipeline with steps 0..DEPTH-1 (32 async ops) ----
    #pragma unroll
    for (int d = 0; d < DEPTH; ++d) {
        const uint64_t sb = base + (uint64_t)d * STEPB;
        async_load_b32(l0 + d * lslot,     off_s0, sb);
        async_load_b32(l0 + d * lslot + 4, off_s1, sb);
    }
    uint64_t pf = base + (uint64_t)DEPTH * STEPB;           // next step to fetch

    float vmem = 0.0f, i0 = 0.0f, i1 = 0.0f;

    // ---- steady state: NGROUP-1 groups of DEPTH steps, invariant 32 in flight
    for (int g = 0; g < NGROUP - 1; ++g) {
        #pragma unroll
        for (int d = 0; d < DEPTH; ++d) {
            wait_async_le<2 * DEPTH - 2>();                 // oldest pair landed

            const float2 x = *reinterpret_cast<const float2*>(&lds[d][tid][0]);
            i0   = fmaf(ALPHA_S0, i0, x.x);                 // leaky synapse 0
            i1   = fmaf(ALPHA_S1, i1, x.y);                 // leaky synapse 1
            vmem = fmaf(ALPHA_MEM, vmem, i0 + i1);          // leaky membrane
            const float sp = (vmem > 1.0f) ? 1.0f : 0.0f;   // threshold
            vmem -= sp;                                     // reset by subtraction
            __builtin_nontemporal_store(sp, Q + (size_t)d * NNEUR);

            // refill slot d with the step DEPTH ahead (uniform SALU base)
            async_load_b32(l0 + d * lslot,     off_s0, pf);
            async_load_b32(l0 + d * lslot + 4, off_s1, pf);
            pf += STEPB;
        }
        Q += (size_t)DEPTH * NNEUR;
    }

    // ---- epilogue: drain once, then consume the final DEPTH steps ----
    wait_async_le<0>();
    #pragma unroll
    for (int d = 0; d < DEPTH; ++d) {
        const float2 x = *reinterpret_cast<const float2*>(&lds[d][tid][0]);
        i0   = fmaf(ALPHA_S0, i0, x.x);
        i1   = fmaf(ALPHA_S1, i1, x.y);
        vmem = fmaf(ALPHA_MEM, vmem, i0 + i1);
        const float sp = (vmem > 1.0f) ? 1.0f : 0.0f;
        vmem -= sp;
        __builtin_nontemporal_store(sp, Q + (size_t)d * NNEUR);
    }
}

extern "C" void kernel_launch(void* const* d_in, const int* in_sizes, int n_in,
                              void* d_out, int out_size, void* d_ws, size_t ws_size,
                              hipStream_t stream) {
    (void)in_sizes; (void)n_in; (void)d_ws; (void)ws_size; (void)out_size;
    const float* in  = (const float*)d_in[0];   // (B*T, NSYN, NNEUR) float32
    float*       out = (float*)d_out;           // (B*T, NNEUR) float32

    // 32 batches * 8 blocks each = 256 blocks of 256 threads (8 wave32s).
    dim3 grid(BATCH * (NNEUR / TPB));
    dim3 block(TPB);
    snn_lif_kernel<<<grid, block, 0, stream>>>(in, out);
}